// AgnosticResidualInteractionBlock_76836964926363
// MI455X (gfx1250) — compile-verified
//
#include <hip/hip_runtime.h>

#define N_NODES 50000
#define N_EDGES 320000
#define CC 64
#define NA 10
#define NR 8
#define WN 320   // 5*C tensor-product weights per edge

// workspace layout (float offsets)
#define OFF_S  (0)                    // s = self-linear scalar   [N,64]
#define OFF_V  (N_NODES * CC)         // v = self-linear vector   [N,64,3]
#define OFF_MS (N_NODES * CC * 4)     // m_s message accumulator  [N,128]
#define OFF_MV (N_NODES * CC * 6)     // m_v message accumulator  [N,192,3]

// d_out layout (float offsets): out_s, out_v, sc_s, sc_v
#define OUT_S   (0)
#define OUT_V   (N_NODES * CC)
#define OUT_SCS (N_NODES * CC * 4)
#define OUT_SCV (N_NODES * CC * 5)

typedef __attribute__((ext_vector_type(2))) float v2f;
typedef __attribute__((ext_vector_type(8))) float v8f;

__device__ __forceinline__ v8f wmma4(v2f a, v2f b, v8f c) {
  // V_WMMA_F32_16X16X4_F32 : D = A(16x4) * B(4x16) + C, all fp32
  return __builtin_amdgcn_wmma_f32_16x16x4_f32(false, a, false, b, (short)0, c,
                                               false, false);
}

__device__ __forceinline__ float silu(float x) {
  return x / (1.0f + __expf(-x));
}

// Async copy of one 16-byte chunk: global (base + goff) -> LDS byte offset.
// Tracked by ASYNCcnt (gfx1250 GLOBAL_LOAD_ASYNC_TO_LDS_B128).
__device__ __forceinline__ void async_copy_b128(unsigned lds_off,
                                                unsigned g_off,
                                                const float* base) {
  asm volatile("global_load_async_to_lds_b128 %0, %1, %2"
               :
               : "v"(lds_off), "v"(g_off), "s"(base)
               : "memory");
}
__device__ __forceinline__ void wait_async0() {
  asm volatile("s_wait_asynccnt 0" ::: "memory");
}

// ---------------------------------------------------------------------------
// Zero the message accumulators (m_s: N*128, m_v: N*576) = 35.2M floats
// ---------------------------------------------------------------------------
__global__ __launch_bounds__(256) void zero_kernel(float4* __restrict__ p) {
  unsigned i = blockIdx.x * 256u + threadIdx.x;
  p[i] = make_float4(0.f, 0.f, 0.f, 0.f);
}

// ---------------------------------------------------------------------------
// Node kernel: skip-connection linear ((feat (x) attr) @ W_sc, K=640) and
// self linear (feat @ W_self, K=64). grid=(N/16, 4); blockIdx.y: 0=scalar,
// 1..3 = vector xyz component. 128 threads = 4 waves = 4 col tiles of 16.
// ---------------------------------------------------------------------------
__global__ __launch_bounds__(128)
void node_kernel(const float* __restrict__ attr, const float* __restrict__ nfs,
                 const float* __restrict__ nfv, const float* __restrict__ Wsc_s,
                 const float* __restrict__ Wsc_v,
                 const float* __restrict__ Wself_s,
                 const float* __restrict__ Wself_v, float* __restrict__ out,
                 float* __restrict__ ws) {
  const int comp = blockIdx.y;
  const int n0 = blockIdx.x << 4;
  const int lane = threadIdx.x & 31;
  const int wave = threadIdx.x >> 5;
  const int row = lane & 15;
  const int hi = lane >> 4;
  const int koff = hi << 1;            // K sub-offset per A/B frag layout
  const int col = (wave << 4) + row;   // output column 0..63

  __shared__ float feat[16][65];       // padded vs bank conflicts
  __shared__ float at[16][17];
  __shared__ float ts[16][644];        // feat (x) attr tile, stride%64==4

  const float* Wsc = (comp == 0) ? Wsc_s : Wsc_v;
  const float* Wself = (comp == 0) ? Wself_s : Wself_v;

  // warm L2 with the weight tile this block streams (global_prefetch_b8)
#pragma unroll
  for (int j = 0; j < 10; ++j)
    __builtin_prefetch(Wsc + (threadIdx.x << 5) + (j << 12), 0, 0);

  for (int i = threadIdx.x; i < 16 * CC; i += 128) {
    int r = i >> 6, c = i & 63;
    feat[r][c] = (comp == 0) ? nfs[(n0 + r) * CC + c]
                             : nfv[((n0 + r) * CC + c) * 3 + (comp - 1)];
  }
  for (int i = threadIdx.x; i < 16 * NA; i += 128) {
    int r = i / NA, a = i - r * NA;
    at[r][a] = attr[(n0 + r) * NA + a];
  }
  __syncthreads();

  // materialize ts[r][c*10+a] = feat[r][c] * attr[r][a]  (once per block)
  for (int i = threadIdx.x; i < 16 * CC * NA; i += 128) {
    int a = i % NA;
    int rest = i / NA;        // = r*64 + c
    int c = rest & 63;
    int r = rest >> 6;
    ts[r][c * NA + a] = feat[r][c] * at[r][a];
  }
  __syncthreads();

  // ---- skip connection GEMM, K = C*NA = 640 ----
  v8f acc = {};
  for (int kk = 0; kk < CC * NA; kk += 4) {
    int k = kk + koff;
    v2f a = {ts[row][k], ts[row][k + 1]};
    v2f b = {Wsc[k * CC + col], Wsc[(k + 1) * CC + col]};
    acc = wmma4(a, b, acc);
  }
  const float inv_sc = 0.0395284708f;  // 1/sqrt(640)
#pragma unroll
  for (int r = 0; r < 8; ++r) {
    int node = n0 + r + (hi << 3);
    float val = acc[r] * inv_sc;
    if (comp == 0)
      __builtin_nontemporal_store(val, &out[OUT_SCS + node * CC + col]);
    else
      __builtin_nontemporal_store(
          val, &out[OUT_SCV + (node * CC + col) * 3 + (comp - 1)]);
  }

  // ---- self linear GEMM, K = 64 ----
  v8f acc2 = {};
  for (int kk = 0; kk < CC; kk += 4) {
    int k = kk + koff;
    v2f a = {feat[row][k], feat[row][k + 1]};
    v2f b = {Wself[k * CC + col], Wself[(k + 1) * CC + col]};
    acc2 = wmma4(a, b, acc2);
  }
#pragma unroll
  for (int r = 0; r < 8; ++r) {
    int node = n0 + r + (hi << 3);
    float val = acc2[r] * 0.125f;      // 1/sqrt(64)
    if (comp == 0) ws[OFF_S + node * CC + col] = val;
    else ws[OFF_V + (node * CC + col) * 3 + (comp - 1)] = val;
  }
}

// ---------------------------------------------------------------------------
// Edge kernel: radial MLP (8->64->64->64->320, WMMA) fused with the
// equivariant tensor product and atomic scatter-add. 16 edges per block,
// 128 threads = 4 waves. The E x 320 weight tensor lives only in LDS.
// ---------------------------------------------------------------------------
__global__ __launch_bounds__(128)
void edge_kernel(const float* __restrict__ sh0, const float* __restrict__ sh1,
                 const float* __restrict__ efeat, const int* __restrict__ snd,
                 const int* __restrict__ rcv, const float* __restrict__ W1,
                 const float* __restrict__ W2, const float* __restrict__ W3,
                 const float* __restrict__ W4, float* __restrict__ ws) {
  const int e0 = blockIdx.x << 4;
  const int lane = threadIdx.x & 31;
  const int wave = threadIdx.x >> 5;
  const int row = lane & 15;
  const int hi = lane >> 4;
  const int koff = hi << 1;
  const int col = (wave << 4) + row;

  __shared__ __align__(16) float efl[16][8];   // contiguous: async target
  __shared__ float hA[16][65];
  __shared__ float hB[16][65];
  __shared__ float wE[16][WN + 1];
  __shared__ float esh[16][4];                 // sh0, sh1.x, sh1.y, sh1.z
  __shared__ int esn[16], ern[16];

  // prefetch the weights this block streams
  __builtin_prefetch(W1 + (threadIdx.x << 2), 0, 0);
  __builtin_prefetch(W2 + (threadIdx.x << 5), 0, 0);
  __builtin_prefetch(W3 + (threadIdx.x << 5), 0, 0);
#pragma unroll
  for (int j = 0; j < 5; ++j)
    __builtin_prefetch(W4 + (threadIdx.x << 5) + (j << 12), 0, 0);

  // async-copy the 512B edge-feature tile straight into LDS (ASYNCcnt path)
  if (threadIdx.x < 32) {
    unsigned lds_off = (unsigned)(uintptr_t)(&efl[0][0]) + threadIdx.x * 16u;
    unsigned g_off = threadIdx.x * 16u;
    async_copy_b128(lds_off, g_off, efeat + (size_t)e0 * NR);
  }
  // stage per-edge scalars for the TP phase
  if (threadIdx.x >= 32 && threadIdx.x < 48) {
    int el = threadIdx.x - 32;
    int e = e0 + el;
    esh[el][0] = sh0[e];
    esh[el][1] = sh1[e * 3 + 0];
    esh[el][2] = sh1[e * 3 + 1];
    esh[el][3] = sh1[e * 3 + 2];
    esn[el] = snd[e];
    ern[el] = rcv[e];
  }
  wait_async0();
  __syncthreads();

  {  // layer 1: K=8, scale 1/sqrt(8), silu
    v8f acc = {};
    for (int kk = 0; kk < NR; kk += 4) {
      int k = kk + koff;
      v2f a = {efl[row][k], efl[row][k + 1]};
      v2f b = {W1[k * 64 + col], W1[(k + 1) * 64 + col]};
      acc = wmma4(a, b, acc);
    }
#pragma unroll
    for (int r = 0; r < 8; ++r)
      hA[r + (hi << 3)][col] = silu(acc[r] * 0.35355339f);
  }
  __syncthreads();
  {  // layer 2: K=64, scale 1/8, silu
    v8f acc = {};
    for (int kk = 0; kk < 64; kk += 4) {
      int k = kk + koff;
      v2f a = {hA[row][k], hA[row][k + 1]};
      v2f b = {W2[k * 64 + col], W2[(k + 1) * 64 + col]};
      acc = wmma4(a, b, acc);
    }
#pragma unroll
    for (int r = 0; r < 8; ++r)
      hB[r + (hi << 3)][col] = silu(acc[r] * 0.125f);
  }
  __syncthreads();
  {  // layer 3: K=64, scale 1/8, silu
    v8f acc = {};
    for (int kk = 0; kk < 64; kk += 4) {
      int k = kk + koff;
      v2f a = {hB[row][k], hB[row][k + 1]};
      v2f b = {W3[k * 64 + col], W3[(k + 1) * 64 + col]};
      acc = wmma4(a, b, acc);
    }
#pragma unroll
    for (int r = 0; r < 8; ++r)
      hA[r + (hi << 3)][col] = silu(acc[r] * 0.125f);
  }
  __syncthreads();
  // layer 4: [16x64] @ [64x320] -> per-edge TP weights (5 col tiles per wave)
  for (int t = 0; t < 5; ++t) {
    int c4 = ((wave + (t << 2)) << 4) + row;
    v8f acc = {};
    for (int kk = 0; kk < 64; kk += 4) {
      int k = kk + koff;
      v2f a = {hA[row][k], hA[row][k + 1]};
      v2f b = {W4[k * WN + c4], W4[(k + 1) * WN + c4]};
      acc = wmma4(a, b, acc);
    }
#pragma unroll
    for (int r = 0; r < 8; ++r)
      wE[r + (hi << 3)][c4] = acc[r] * 0.125f;
  }
  __syncthreads();

  // ---- tensor product + scatter-add (1024 (edge,channel) items / block) ----
  for (int it = 0; it < 8; ++it) {
    int idx = threadIdx.x + (it << 7);
    int el = idx >> 6;
    int c = idx & 63;
    int sN = esn[el];
    int rN = ern[el];
    float s0 = esh[el][0];
    float sx = esh[el][1], sy = esh[el][2], sz = esh[el][3];
    float se = ws[OFF_S + sN * CC + c];
    const float* vp = ws + OFF_V + (sN * CC + c) * 3;
    float vx = vp[0], vy = vp[1], vz = vp[2];
    float w1v = wE[el][c], w2v = wE[el][64 + c], w3v = wE[el][128 + c];
    float w4v = wE[el][192 + c], w5v = wE[el][256 + c];
    float p1 = w1v * se * s0;                                        // 0e x 0e
    float p2 = w2v * (vx * sx + vy * sy + vz * sz) * 0.57735027f;    // 1o.1o
    float a3 = w3v * se;                                             // 0e x 1o
    float a4 = w4v * s0;                                             // 1o x 0e
    float k5 = w5v * 0.70710678f;                                    // 1o x 1o
    float cx = vy * sz - vz * sy;
    float cy = vz * sx - vx * sz;
    float cz = vx * sy - vy * sx;
    float* ms = ws + OFF_MS + rN * (2 * CC);
    unsafeAtomicAdd(ms + c, p1);
    unsafeAtomicAdd(ms + CC + c, p2);
    float* mv = ws + OFF_MV + rN * (3 * CC) * 3;
    unsafeAtomicAdd(mv + c * 3 + 0, a3 * sx);
    unsafeAtomicAdd(mv + c * 3 + 1, a3 * sy);
    unsafeAtomicAdd(mv + c * 3 + 2, a3 * sz);
    unsafeAtomicAdd(mv + (CC + c) * 3 + 0, a4 * vx);
    unsafeAtomicAdd(mv + (CC + c) * 3 + 1, a4 * vy);
    unsafeAtomicAdd(mv + (CC + c) * 3 + 2, a4 * vz);
    unsafeAtomicAdd(mv + (2 * CC + c) * 3 + 0, k5 * cx);
    unsafeAtomicAdd(mv + (2 * CC + c) * 3 + 1, k5 * cy);
    unsafeAtomicAdd(mv + (2 * CC + c) * 3 + 2, k5 * cz);
  }
}

// ---------------------------------------------------------------------------
// Output linear: out_s = m_s @ W_out_s (K=128), out_v = m_v @ W_out_v (K=192)
// grid=(N/16, 4); blockIdx.y: 0=scalar, 1..3=vector component.
// The contiguous m_s tile is staged via GLOBAL_LOAD_ASYNC_TO_LDS_B128.
// ---------------------------------------------------------------------------
#define ATS 196  // Atile row stride (floats): %64==4 (no bank conflicts), 16B-aligned rows
__global__ __launch_bounds__(128)
void out_kernel(const float* __restrict__ Wout_s,
                const float* __restrict__ Wout_v, const float* __restrict__ ws,
                float* __restrict__ out) {
  const int comp = blockIdx.y;
  const int n0 = blockIdx.x << 4;
  const int lane = threadIdx.x & 31;
  const int wave = threadIdx.x >> 5;
  const int row = lane & 15;
  const int hi = lane >> 4;
  const int koff = hi << 1;
  const int col = (wave << 4) + row;

  __shared__ __align__(16) float Atile[16][ATS];

  const float* W = (comp == 0) ? Wout_s : Wout_v;
  __builtin_prefetch(W + (threadIdx.x << 5), 0, 0);
  __builtin_prefetch(W + (threadIdx.x << 5) + 4096, 0, 0);
  __builtin_prefetch(W + (threadIdx.x << 5) + 8192, 0, 0);

  const int K = (comp == 0) ? 2 * CC : 3 * CC;
  if (comp == 0) {
    // 16 rows x 128 floats contiguous per row: 512 x 16B async chunks
    const float* base = ws + OFF_MS + (size_t)n0 * (2 * CC);
    unsigned abase = (unsigned)(uintptr_t)(&Atile[0][0]);
    for (int t = threadIdx.x; t < 512; t += 128) {
      int r = t >> 5;    // row
      int ch = t & 31;   // 16B chunk within row
      unsigned lds_off = abase + (unsigned)(r * (ATS * 4) + (ch << 4));
      unsigned g_off = (unsigned)((r * (2 * CC) + (ch << 2)) << 2);
      async_copy_b128(lds_off, g_off, base);
    }
    wait_async0();
  } else {
    for (int i = threadIdx.x; i < 16 * 3 * CC; i += 128) {
      int r = i / (3 * CC), k = i - r * (3 * CC);
      Atile[r][k] = ws[OFF_MV + ((n0 + r) * (3 * CC) + k) * 3 + (comp - 1)];
    }
  }
  __syncthreads();

  v8f acc = {};
  for (int kk = 0; kk < K; kk += 4) {
    int k = kk + koff;
    v2f a = {Atile[row][k], Atile[row][k + 1]};
    v2f b = {W[k * CC + col], W[(k + 1) * CC + col]};
    acc = wmma4(a, b, acc);
  }
  const float scale = (comp == 0) ? 0.0055242717f   // 1/(sqrt(128)*16)
                                  : 0.0045105654f;  // 1/(sqrt(192)*16)
#pragma unroll
  for (int r = 0; r < 8; ++r) {
    int node = n0 + r + (hi << 3);
    float val = acc[r] * scale;
    if (comp == 0)
      __builtin_nontemporal_store(val, &out[OUT_S + node * CC + col]);
    else
      __builtin_nontemporal_store(
          val, &out[OUT_V + (node * CC + col) * 3 + (comp - 1)]);
  }
}

extern "C" void kernel_launch(void* const* d_in, const int* in_sizes, int n_in,
                              void* d_out, int out_size, void* d_ws,
                              size_t ws_size, hipStream_t stream) {
  const float* attr = (const float*)d_in[0];
  const float* nfs = (const float*)d_in[1];
  const float* nfv = (const float*)d_in[2];
  const float* sh0 = (const float*)d_in[3];
  const float* sh1 = (const float*)d_in[4];
  const float* efeat = (const float*)d_in[5];
  const int* snd = (const int*)d_in[6];
  const int* rcv = (const int*)d_in[7];
  const float* Wsc_s = (const float*)d_in[8];
  const float* Wsc_v = (const float*)d_in[9];
  const float* Wself_s = (const float*)d_in[10];
  const float* Wself_v = (const float*)d_in[11];
  const float* W1 = (const float*)d_in[12];
  const float* W2 = (const float*)d_in[13];
  const float* W3 = (const float*)d_in[14];
  const float* W4 = (const float*)d_in[15];
  const float* Wout_s = (const float*)d_in[16];
  const float* Wout_v = (const float*)d_in[17];
  float* out = (float*)d_out;
  float* ws = (float*)d_ws;

  // m_s + m_v = N*(128+576) = 35,200,000 floats = 8,800,000 float4
  zero_kernel<<<34375, 256, 0, stream>>>((float4*)(ws + OFF_MS));
  dim3 gn(N_NODES / 16, 4);
  node_kernel<<<gn, 128, 0, stream>>>(attr, nfs, nfv, Wsc_s, Wsc_v, Wself_s,
                                      Wself_v, out, ws);
  edge_kernel<<<N_EDGES / 16, 128, 0, stream>>>(sh0, sh1, efeat, snd, rcv, W1,
                                                W2, W3, W4, ws);
  out_kernel<<<gn, 128, 0, stream>>>(Wout_s, Wout_v, ws, out);
}